// MultiHeadAttention_88192858456426
// MI455X (gfx1250) — compile-verified
//
#include <hip/hip_runtime.h>
#include <hip/hip_bf16.h>

typedef __attribute__((ext_vector_type(2))) float v2f;
typedef __attribute__((ext_vector_type(8))) float v8f;

static constexpr int Bn = 8;     // batch
static constexpr int Cd = 512;   // channels
static constexpr int Nd = 2048;  // tokens

#define WMMA_F32(a, b, c) \
    __builtin_amdgcn_wmma_f32_16x16x4_f32(false, (a), false, (b), (short)0, (c), false, false)

// ---------------------------------------------------------------------------
// Projection, output layout [C, N]:  P[c,n] = sum_i W[c,i] * X[i,n] + bias[c]
// Block: 256 thr (8 waves). Tile: 16 rows (c) x 128 cols (n), wave = 16x16.
// ---------------------------------------------------------------------------
__global__ __launch_bounds__(256) void proj_cn_kernel(
    const float* __restrict__ X,     // [B, C, N]
    const float* __restrict__ W,     // [C, C]
    const float* __restrict__ bias,  // [C]
    float* __restrict__ P)           // [B, C, N]
{
    __shared__ float wt[16][516];    // padded: stride 516 -> conflict-free col reads

    const int tid  = threadIdx.x;
    const int wave = tid >> 5;
    const int lane = tid & 31;
    const int l15  = lane & 15;
    const int hi2  = (lane >> 4) * 2;   // K sub-offset for A/B frags
    const int rofs = (lane >> 4) * 8;   // D-matrix row offset for upper lanes

    const int n0 = blockIdx.x * 128;
    const int c0 = blockIdx.y * 16;
    const int b  = blockIdx.z;

    const float* Xb = X + (size_t)b * Cd * Nd;

    // cooperative load of W rows [c0, c0+16), all 512 cols (coalesced)
#pragma unroll
    for (int j = 0; j < 32; ++j) {
        int idx = j * 256 + tid;
        int r   = idx >> 9;
        int col = idx & 511;
        wt[r][col] = W[(size_t)(c0 + r) * Cd + col];
    }
    __syncthreads();

    const int ncol = n0 + wave * 16 + l15;
    v8f acc = {};
#pragma unroll 8
    for (int k0 = 0; k0 < Cd; k0 += 4) {
        const int ka = k0 + hi2;
        v2f a = *(const v2f*)&wt[l15][ka];          // A[c, k] from LDS (8B aligned)
        const float* xp = Xb + (size_t)ka * Nd + ncol;
        v2f bf;
        bf.x = xp[0];                               // B[k, n] lane-consecutive n
        bf.y = xp[Nd];
        acc = WMMA_F32(a, bf, acc);
    }

    float* Pb = P + (size_t)b * Cd * Nd;
#pragma unroll
    for (int r = 0; r < 8; ++r) {
        const int c = c0 + r + rofs;
        Pb[(size_t)c * Nd + ncol] = acc[r] + bias[c];
    }
}

// ---------------------------------------------------------------------------
// Projection, output layout [N, C]:  P[n,c] = sum_i X[i,n] * W[c,i] + bias[c]
// Block: 256 thr. Tile: 16 rows (n) x 128 cols (c), wave = 16x16.
// ---------------------------------------------------------------------------
__global__ __launch_bounds__(256) void proj_nc_kernel(
    const float* __restrict__ X,     // [B, C, N]
    const float* __restrict__ W,     // [C, C]
    const float* __restrict__ bias,  // [C]
    float* __restrict__ P)           // [B, N, C]
{
    __shared__ float wt[128][33];    // odd stride -> conflict-free

    const int tid  = threadIdx.x;
    const int wave = tid >> 5;
    const int lane = tid & 31;
    const int l15  = lane & 15;
    const int hi2  = (lane >> 4) * 2;
    const int rofs = (lane >> 4) * 8;

    const int c0 = blockIdx.x * 128;
    const int n0 = blockIdx.y * 16;
    const int b  = blockIdx.z;

    const float* Xb = X + (size_t)b * Cd * Nd;
    const int cw = c0 + wave * 16 + l15;   // this lane's output column (c)

    v8f acc = {};
    for (int k0 = 0; k0 < Cd; k0 += 32) {
        __syncthreads();
        // stage W[c0..c0+128), k0..k0+32) into LDS
#pragma unroll
        for (int j = 0; j < 16; ++j) {
            int idx = j * 256 + tid;
            int c = idx >> 5, kk = idx & 31;
            wt[c][kk] = W[(size_t)(c0 + c) * Cd + k0 + kk];
        }
        __syncthreads();
#pragma unroll
        for (int k = 0; k < 32; k += 4) {
            const int ka = k0 + k + hi2;
            const float* xp = Xb + (size_t)ka * Nd + n0 + l15;
            v2f a;
            a.x = xp[0];                            // A[n, k] lane-consecutive n
            a.y = xp[Nd];
            const int kl = k + hi2;
            v2f bf;
            bf.x = wt[wave * 16 + l15][kl];         // B[k, c] from LDS
            bf.y = wt[wave * 16 + l15][kl + 1];
            acc = WMMA_F32(a, bf, acc);
        }
    }

    float* Pb = P + (size_t)b * Nd * Cd;
#pragma unroll
    for (int r = 0; r < 8; ++r) {
        const int n = n0 + r + rofs;
        Pb[(size_t)n * Cd + cw] = acc[r] + bias[cw];
    }
}

// ---------------------------------------------------------------------------
// Fused attention: for a 16-row n-tile, loop m-tiles of 16:
//   S = Qp_tile * Kp_tile^T   (reduce over C=512, split across 8 waves + LDS)
//   O += S * Vp_tile          (each wave owns 64 of the 512 O columns)
// Epilogue: transpose O via LDS, add residual, store [C, N] coalesced.
// ---------------------------------------------------------------------------
__global__ __launch_bounds__(256) void attn_fused_kernel(
    const float* __restrict__ Qp,    // [B, C, N]
    const float* __restrict__ Kp,    // [B, C, N]
    const float* __restrict__ Vp,    // [B, N, C]
    const float* __restrict__ Qres,  // original q [B, C, N]
    float* __restrict__ Out)         // [B, C, N]
{
    __shared__ float qt[512][16];    // Q tile [c'][n_local]; reused for O staging
    __shared__ float spart[8][256];  // per-wave partial S (16x16 each)
    __shared__ float sred[16][17];   // reduced S, padded

    const int tid  = threadIdx.x;
    const int wave = tid >> 5;
    const int lane = tid & 31;
    const int l15  = lane & 15;
    const int hi2  = (lane >> 4) * 2;
    const int rofs = (lane >> 4) * 8;

    const int n0 = blockIdx.x * 16;
    const int b  = blockIdx.y;

    const float* Qpb = Qp + (size_t)b * Cd * Nd;
    const float* Kpb = Kp + (size_t)b * Cd * Nd;
    const float* Vpb = Vp + (size_t)b * Nd * Cd;

    // stage Q tile: qt[c'][n_local]
#pragma unroll
    for (int j = 0; j < 32; ++j) {
        int idx = j * 256 + tid;
        int c = idx >> 4, nl = idx & 15;
        qt[c][nl] = Qpb[(size_t)c * Nd + n0 + nl];
    }
    __syncthreads();

    const int c0w = wave * 64;       // this wave's 64-channel chunk
    v8f acc[4] = {};                 // O columns [c0w, c0w+64)

    for (int m0 = 0; m0 < Nd; m0 += 16) {
        // ---- GEMM1: partial S over this wave's channel chunk (16 WMMAs) ----
        v8f sp = {};
#pragma unroll
        for (int k = 0; k < 64; k += 4) {
            const int ka = c0w + k + hi2;
            v2f a;
            a.x = qt[ka][l15];
            a.y = qt[ka + 1][l15];
            const float* kp = Kpb + (size_t)ka * Nd + m0 + l15;
            v2f bf;
            bf.x = kp[0];
            bf.y = kp[Nd];
            sp = WMMA_F32(a, bf, sp);
        }
#pragma unroll
        for (int r = 0; r < 8; ++r)
            spart[wave][(r + rofs) * 16 + l15] = sp[r];
        __syncthreads();

        // ---- cross-wave reduction of S ----
        {
            float s = 0.f;
#pragma unroll
            for (int w = 0; w < 8; ++w) s += spart[w][tid];
            sred[tid >> 4][tid & 15] = s;
        }
        __syncthreads();

        // ---- GEMM2: O += S * V  (4 col-subtiles x 4 k-steps = 16 WMMAs) ----
#pragma unroll
        for (int k = 0; k < 16; k += 4) {
            const int km = k + hi2;
            v2f a;
            a.x = sred[l15][km];
            a.y = sred[l15][km + 1];
            const float* vrow = Vpb + (size_t)(m0 + km) * Cd + c0w + l15;
            v2f b0, b1, b2, b3;
            b0.x = vrow[0];      b0.y = vrow[Cd];
            b1.x = vrow[16];     b1.y = vrow[Cd + 16];
            b2.x = vrow[32];     b2.y = vrow[Cd + 32];
            b3.x = vrow[48];     b3.y = vrow[Cd + 48];
            acc[0] = WMMA_F32(a, b0, acc[0]);
            acc[1] = WMMA_F32(a, b1, acc[1]);
            acc[2] = WMMA_F32(a, b2, acc[2]);
            acc[3] = WMMA_F32(a, b3, acc[3]);
        }
        __syncthreads();   // protect spart/sred for next iteration
    }

    // ---- epilogue: stage O into qt (transpose), then coalesced store ----
#pragma unroll
    for (int s = 0; s < 4; ++s)
#pragma unroll
        for (int r = 0; r < 8; ++r)
            qt[c0w + s * 16 + l15][r + rofs] = acc[s][r];
    __syncthreads();

    const float* Qr = Qres + (size_t)b * Cd * Nd;
    float* Ob = Out + (size_t)b * Cd * Nd;
#pragma unroll
    for (int j = 0; j < 32; ++j) {
        int idx = j * 256 + tid;
        int c = idx >> 4, nl = idx & 15;
        size_t o = (size_t)c * Nd + n0 + nl;
        Ob[o] = qt[c][nl] + Qr[o];   // residual add, 64B-coalesced
    }
}

// ---------------------------------------------------------------------------
extern "C" void kernel_launch(void* const* d_in, const int* in_sizes, int n_in,
                              void* d_out, int out_size, void* d_ws, size_t ws_size,
                              hipStream_t stream) {
    const float* q  = (const float*)d_in[0];
    const float* k  = (const float*)d_in[1];
    const float* v  = (const float*)d_in[2];
    const float* wq = (const float*)d_in[3];
    const float* bq = (const float*)d_in[4];
    const float* wk = (const float*)d_in[5];
    const float* bk = (const float*)d_in[6];
    const float* wv = (const float*)d_in[7];
    const float* bv = (const float*)d_in[8];
    float* out = (float*)d_out;

    float* ws = (float*)d_ws;
    const size_t per = (size_t)Bn * Cd * Nd;
    float* qp = ws;             // [B, C, N]
    float* kp = ws + per;       // [B, C, N]
    float* vp = ws + 2 * per;   // [B, N, C]

    dim3 blk(256);
    proj_cn_kernel<<<dim3(Nd / 128, Cd / 16, Bn), blk, 0, stream>>>(q, wq, bq, qp);
    proj_cn_kernel<<<dim3(Nd / 128, Cd / 16, Bn), blk, 0, stream>>>(k, wk, bk, kp);
    proj_nc_kernel<<<dim3(Cd / 128, Nd / 16, Bn), blk, 0, stream>>>(v, wv, bv, vp);
    attn_fused_kernel<<<dim3(Nd / 16, Bn), blk, 0, stream>>>(qp, kp, vp, q, out);
}